// Simplified_MultiHeadAttention_45226005627418
// MI455X (gfx1250) — compile-verified
//
#include <hip/hip_runtime.h>

typedef __attribute__((ext_vector_type(2))) float v2f;
typedef __attribute__((ext_vector_type(4))) float v4f;
typedef __attribute__((ext_vector_type(8))) float v8f;
typedef __attribute__((ext_vector_type(4))) unsigned int v4u;
typedef __attribute__((ext_vector_type(4))) int v4i;
typedef __attribute__((ext_vector_type(8))) int v8i;

#define B_    16
#define T_    4096
#define D_    128
#define H_    8
#define TP1   4097
#define OUTC  1152

#define TILE_T 128
#define LROWS  144          // TILE_T + 16-row halo
#define LPITCH 136          // padded LDS row pitch (floats): bank-conflict-free B loads
#define WROW   40           // padded weight-table row: zeros at [0,16) and [24,40)
#define WTAB_OFF (LROWS * LPITCH)              // 19584 floats
#define LDS_FLOATS (WTAB_OFF + 72 * WROW)      // + 72*40 = 22464 floats (~89.9 KB)

#if __has_builtin(__builtin_amdgcn_tensor_load_to_lds)
#define HAVE_TDM 1
#else
#define HAVE_TDM 0
#endif

// ---------------- full-tile kernel: blocks 0..31, every t in range ----------------
__launch_bounds__(256)
__global__ void mha_conv_full(const float* __restrict__ X,
                              const float* __restrict__ W,
                              float* __restrict__ out)
{
    __shared__ float lds[LDS_FLOATS];
    const int tid = threadIdx.x;
    const int b  = blockIdx.y;
    const int t0 = blockIdx.x * TILE_T;

    // staged rows: [gstart, t0+128) -> LDS rows [rdst, 144);  rdst = 16 only for block 0
    const int gstart = (t0 - 16) < 0 ? 0 : (t0 - 16);
    const int nrows  = (t0 + TILE_T) - gstart;
    const int rdst   = gstart - (t0 - 16);

#if HAVE_TDM
    // ---- 1a) TDM: DMA X rows -> LDS; 512B rows + 32B hw pad = 136-float pitch
    if (tid < 32) {
        const unsigned long long gaddr =
            (unsigned long long)(const void*)(X + (size_t)(b * T_ + gstart) * D_);
        const unsigned int ldsb = (unsigned int)(size_t)(const void*)&lds[rdst * LPITCH];

        const v4u g0 = {
            1u,                                                       // count=1, user descriptor
            ldsb,                                                     // lds_addr (bytes)
            (unsigned int)(gaddr & 0xffffffffu),                      // global_addr[31:0]
            (unsigned int)((gaddr >> 32) & 0x01ffffffu) | (2u << 30)  // addr[56:32] | type=2
        };
        const v8i g1 = {
            (2 << 16) | (1 << 20) | (6 << 22) | (7 << 25),  // 4B elems, pad_en, 128dw interval, 8dw pad
            (int)((D_ & 0xffff) << 16),                     // tensor_dim0[15:0]=128
            (int)((unsigned)nrows << 16),                   // tensor_dim0 hi=0 | tensor_dim1 lo=nrows
            (int)(D_ << 16),                                // tensor_dim1 hi=0 | tile_dim0=128
            (int)(nrows & 0xffff),                          // tile_dim1=nrows, tile_dim2=0
            D_,                                             // tensor_dim0_stride = 128 elems
            0, 0
        };
        const v4i z4 = {0, 0, 0, 0};
#if __clang_major__ >= 23
        const v8i z8 = {0, 0, 0, 0, 0, 0, 0, 0};
        __builtin_amdgcn_tensor_load_to_lds(g0, g1, z4, z4, z8, 0);
#else
        __builtin_amdgcn_tensor_load_to_lds(g0, g1, z4, z4, 0);
#endif
    }
#endif

    // ---- 2) softmax weight table (overlaps DMA): padded rows, p[j] at offset 16+j
    if (tid < 72) {
        const int h  = tid / 9;
        const int nv = tid % 9;
        float e[H_];
        float mx = -3.0e38f;
        for (int j = 0; j < H_; ++j) {
            float wv = W[(h + j) * H_ + h];
            e[j] = wv;
            if (j < nv && wv > mx) mx = wv;
        }
        float s = 0.f;
        for (int j = 0; j < H_; ++j) {
            float v = (j < nv) ? expf(e[j] - mx) : 0.f;
            e[j] = v;
            s += v;
        }
        const float inv = (nv > 0) ? (1.f / fmaxf(s, 1e-30f)) : 0.f;
        for (int jj = 0; jj < WROW; ++jj)
            lds[WTAB_OFF + tid * WROW + jj] =
                (jj >= 16 && jj < 16 + H_) ? e[jj - 16] * inv : 0.f;
    }

#if HAVE_TDM
    // ---- 1b) zero-fill halo rows not covered by the DMA (block 0 only)
    if (rdst > 0) {
        #pragma unroll
        for (int it = 0; it < 2; ++it) {
            const int idx = tid + it * 256;        // 16 rows * 32 float4
            const int r   = idx >> 5;
            const int c4  = (idx & 31) << 2;
            *reinterpret_cast<v4f*>(&lds[r * LPITCH + c4]) = (v4f){0.f, 0.f, 0.f, 0.f};
        }
    }
#if __has_builtin(__builtin_amdgcn_s_wait_tensorcnt)
    __builtin_amdgcn_s_wait_tensorcnt(0);
#else
    asm volatile("s_wait_tensorcnt 0x0" ::: "memory");
#endif
#else
    // ---- fallback staging: vectorized global->LDS loads, zero-fill OOB
    {
        #pragma unroll
        for (int it = 0; it < 18; ++it) {
            const int idx = tid + it * 256;
            const int r   = idx >> 5;
            const int c4  = (idx & 31) << 2;
            const int g   = t0 - 16 + r;
            v4f v = {0.f, 0.f, 0.f, 0.f};
            if (g >= 0)
                v = *reinterpret_cast<const v4f*>(X + (size_t)(b * T_ + g) * D_ + c4);
            *reinterpret_cast<v4f*>(&lds[r * LPITCH + c4]) = v;
        }
    }
#endif
    __syncthreads();

    // ---- 3) output chunk 0: copy of X (LDS -> global, b128, unguarded)
    {
        #pragma unroll
        for (int it = 0; it < 16; ++it) {
            const int idx = tid + it * 256;        // 128 rows * 32 float4
            const int r   = idx >> 5;
            const int c4  = (idx & 31) << 2;
            v4f v = *reinterpret_cast<const v4f*>(&lds[(r + 16) * LPITCH + c4]);
            *reinterpret_cast<v4f*>(out + ((size_t)b * TP1 + t0 + r) * OUTC + c4) = v;
        }
    }

    // ---- 4) banded-Toeplitz x X via f32 WMMA: D(16x16) += A(16x4) * B(4x16)
    const int lane = tid & 31;
    const int wsub = tid >> 5;           // wave id == t-subtile 0..7
    const int i    = lane & 15;          // A: row M;   B: col N
    const int kh   = lane >> 4;          // half-wave -> K pair select
    const int tstart = t0 + wsub * 16;

    for (int h = 0; h < H_; ++h) {
        int nv = tstart + i - h;
        nv = nv < 0 ? 0 : (nv > 8 ? 8 : nv);
        // centered pointer: wtp[j] valid (zero-padded) for j in [-16, 23]
        const float* wtp = &lds[WTAB_OFF + (h * 9 + nv) * WROW + 16];

        v2f a[6];
        #pragma unroll
        for (int kc = 0; kc < 6; ++kc) {
            const int j0 = i + 7 - (kc * 4 + kh * 2);   // in [-15, 22]
            a[kc].x = wtp[j0];
            a[kc].y = wtp[j0 - 1];
        }

        const int lrow0 = wsub * 16 + 8 - h;

        for (int ds = 0; ds < 8; ++ds) {
            const int col = ds * 16 + i;
            v8f c = {0.f, 0.f, 0.f, 0.f, 0.f, 0.f, 0.f, 0.f};
            #pragma unroll
            for (int kc = 0; kc < 6; ++kc) {
                const int lr = lrow0 + kc * 4 + kh * 2;
                v2f bm;
                bm.x = lds[lr * LPITCH + col];
                bm.y = lds[(lr + 1) * LPITCH + col];
                c = __builtin_amdgcn_wmma_f32_16x16x4_f32(false, a[kc], false, bm,
                                                          (short)0, c, false, false);
            }
            // C/D layout: VGPR v -> row M = v + 8*kh, col N = i
            const size_t obase = ((size_t)b * TP1) * OUTC + (size_t)(h + 1) * D_ + ds * 16 + i;
            #pragma unroll
            for (int v = 0; v < 8; ++v) {
                const int t = tstart + v + 8 * kh;
                out[obase + (size_t)t * OUTC] = c[v];
            }
        }
    }
}

// ---------------- tail kernel: only row t = 4096 ----------------
__launch_bounds__(128)
__global__ void mha_conv_tail(const float* __restrict__ X,
                              const float* __restrict__ W,
                              float* __restrict__ out)
{
    __shared__ float wt[H_][H_];
    const int b = blockIdx.x;
    const int d = threadIdx.x;          // 128 threads, one per feature

    if (threadIdx.x < H_) {
        const int h = threadIdx.x;
        float e[H_];
        float mx = -3.0e38f;
        for (int j = 0; j < H_; ++j) { e[j] = W[(h + j) * H_ + h]; mx = fmaxf(mx, e[j]); }
        float s = 0.f;
        for (int j = 0; j < H_; ++j) { e[j] = expf(e[j] - mx); s += e[j]; }
        const float inv = 1.f / fmaxf(s, 1e-30f);
        for (int j = 0; j < H_; ++j) wt[h][j] = e[j] * inv;
    }
    __syncthreads();

    float* orow = out + ((size_t)b * TP1 + T_) * OUTC;
    orow[d] = 0.f;                      // chunk 0: X_tilde row T is zeros
    for (int h = 0; h < H_; ++h) {      // t=4096 => nv=8, sources rows 4095-h-j
        float acc = 0.f;
        #pragma unroll
        for (int j = 0; j < H_; ++j)
            acc += wt[h][j] * X[((size_t)b * T_ + (T_ - 1 - h - j)) * D_ + d];
        orow[(h + 1) * D_ + d] = acc;
    }
}

extern "C" void kernel_launch(void* const* d_in, const int* in_sizes, int n_in,
                              void* d_out, int out_size, void* d_ws, size_t ws_size,
                              hipStream_t stream) {
    (void)in_sizes; (void)n_in; (void)d_ws; (void)ws_size; (void)out_size;
    const float* X = (const float*)d_in[0];   // (16, 4096, 128) f32
    const float* W = (const float*)d_in[1];   // (4096, 8) f32
    float* out = (float*)d_out;               // (16, 4097, 1152) f32

    dim3 grid(T_ / TILE_T, B_);               // 32 x 16, all fully in range
    mha_conv_full<<<grid, 256, 0, stream>>>(X, W, out);
    mha_conv_tail<<<B_, 128, 0, stream>>>(X, W, out);
}